// GAT_46213848105786
// MI455X (gfx1250) — compile-verified
//
#include <hip/hip_runtime.h>
#include <hip/hip_bf16.h>
#include <math.h>

// ---- problem constants (from reference) ----
#define IN_DIM  128
#define HID1    64
#define HEADS1  64
#define HEADS2  32
#define NCLS    32
#define HC1     (HEADS1 * HID1)   // 4096
#define HC2     (HEADS2 * NCLS)   // 1024

typedef __attribute__((ext_vector_type(2))) float v2f;
typedef __attribute__((ext_vector_type(8))) float v8f;

// LDS staging geometry for the GEMM B panel
#define KCHUNK       16                  // k-rows per LDS chunk
#define LDSW         68                  // padded row stride (floats) - kills 2-way bank conflicts
#define CHUNK_FLOATS (KCHUNK * LDSW)     // 1088
#define CHUNK_BYTES  (CHUNK_FLOATS * 4)  // 4352 (row pitch 272 = 17*16 -> b128-aligned stores)
#define GEMM_LDS     (2 * CHUNK_BYTES)   // double buffered

// Async copy of 16B global -> LDS, tracked by ASYNCcnt (CDNA5 §10.7).
// LDS byte offset comes from the vdst VGPR, global address from vaddr (no SADDR).
__device__ __forceinline__ void async_b128_to_lds(unsigned int lds_byte_off,
                                                  const float* gaddr) {
  asm volatile("global_load_async_to_lds_b128 %0, %1, off"
               :: "v"(lds_byte_off), "v"((unsigned long long)(uintptr_t)gaddr)
               : "memory");
}

// ---------------------------------------------------------------------------
// C[M x N] = A[M x K] @ B[K x N], fp32, V_WMMA_F32_16X16X4_F32 (CDNA5).
// Block = 4 waves, 64x64 C tile; wave w owns rows [tm*64+16w, +16).
// B panel (64 cols x K) is shared by all 4 waves, staged through LDS with
// double-buffered global_load_async_to_lds_b128 + s_wait_asynccnt.
// WMMA VGPR layouts per ISA 7.12.2:
//   A 16x4 : lane = 16*half + m ; v0 = A[m][k+2*half], v1 = A[m][k+2*half+1]
//   B 4x16 : lane = 16*half + n ; v0 = B[k+2*half][n], v1 = B[k+2*half+1][n]
//   C 16x16: VGPR v, lanes 0-15 -> M=v, lanes 16-31 -> M=v+8 ; N = lane&15
// Out-of-range M rows are CLAMPED (not masked): duplicate rows only feed
// D rows >= M which are never stored, and EXEC stays all-ones for WMMA.
// ---------------------------------------------------------------------------
template <int K, int N>
__global__ __launch_bounds__(128)
void gat_wmma_gemm_f32(const float* __restrict__ A,
                       const float* __restrict__ B,
                       float* __restrict__ C, int M) {
  extern __shared__ float smem[];          // no static LDS -> offset 0
  const int tid  = threadIdx.x;
  const int wave = tid >> 5;
  const int lane = tid & 31;
  const int half = lane >> 4;              // 0: lanes 0-15, 1: lanes 16-31
  const int l    = lane & 15;
  const int tn   = blockIdx.x;             // 64-column super-tile
  const int tm   = blockIdx.y * 4 + wave;  // this wave's 16-row tile
  int row = tm * 16 + l;
  if (row >= M) row = M - 1;               // clamp instead of predicate
  const int col0 = tn * 64 + l;
  const float* __restrict__ Ap = A + (size_t)row * K + half * 2;
  const float* __restrict__ Bc = B + tn * 64;   // B panel col base

  // loader decomposition: thread tid copies rows (tid>>4) and (tid>>4)+8,
  // 4 consecutive floats starting at col (tid&15)*4, per chunk.
  const int lr = tid >> 4;                 // 0..7
  const int lc = (tid & 15) * 4;           // 0..60

  v8f acc[4] = {};

  // prologue: chunk 0 -> buffer 0
  {
    const float* g = Bc + (size_t)lr * N + lc;
    async_b128_to_lds((unsigned)((lr * LDSW + lc) * 4), g);
    async_b128_to_lds((unsigned)(((lr + 8) * LDSW + lc) * 4), g + (size_t)8 * N);
  }

  const int nc = K / KCHUNK;
  for (int c = 0; c < nc; ++c) {
    const int buf = c & 1;
    if (c + 1 < nc) {                      // issue next chunk (uniform branch)
      const int k0n = (c + 1) * KCHUNK;
      const unsigned base = (unsigned)(((c + 1) & 1) * CHUNK_BYTES);
      const float* g = Bc + (size_t)(k0n + lr) * N + lc;
      async_b128_to_lds(base + (unsigned)((lr * LDSW + lc) * 4), g);
      async_b128_to_lds(base + (unsigned)(((lr + 8) * LDSW + lc) * 4),
                        g + (size_t)8 * N);
      asm volatile("s_wait_asynccnt 0x2" ::: "memory");  // chunk c landed
    } else {
      asm volatile("s_wait_asynccnt 0x0" ::: "memory");
    }
    __syncthreads();                       // chunk c visible to all waves

    const float* __restrict__ sb = smem + buf * CHUNK_FLOATS;
    const float* __restrict__ Apc = Ap + c * KCHUNK;
#pragma unroll
    for (int k = 0; k < KCHUNK; k += 4) {
      v2f a = *(const v2f*)(Apc + k);      // global_load_b64 (per-wave A rows)
      const int r0 = (k + half * 2) * LDSW + l;
      // Load each B fragment directly as the (row kk, row kk+1) pair so the
      // ds_load_2addr_b32 result lands in the VGPR pair WMMA consumes.
      v2f bf[4];
#pragma unroll
      for (int t = 0; t < 4; ++t) {
        bf[t].x = sb[r0 + 16 * t];
        bf[t].y = sb[r0 + LDSW + 16 * t];
      }
#pragma unroll
      for (int t = 0; t < 4; ++t) {
        acc[t] = __builtin_amdgcn_wmma_f32_16x16x4_f32(
            false, a, false, bf[t], (short)0, acc[t], false, false);
      }
    }
    __syncthreads();                       // WAR: done reading before overwrite
  }

  const int crow0 = tm * 16 + half * 8;
#pragma unroll
  for (int v = 0; v < 8; ++v) {
    const int crow = crow0 + v;
    if (crow < M) {
#pragma unroll
      for (int t = 0; t < 4; ++t)
        C[(size_t)crow * N + col0 + 16 * t] = acc[t][v];
    }
  }
}

// ---------------------------------------------------------------------------
// a_src[n,h] = dot(h[n,h,:], att_src[h,:]) ; same for a_dst. Thread per (n,h).
// ---------------------------------------------------------------------------
template <int H, int C>
__global__ void gat_att_dots(const float* __restrict__ h,
                             const float* __restrict__ att_src,
                             const float* __restrict__ att_dst,
                             float* __restrict__ asrc,
                             float* __restrict__ adst,
                             int N) {
  int idx = blockIdx.x * blockDim.x + threadIdx.x;
  if (idx >= N * H) return;
  int n = idx / H, hd = idx & (H - 1);
  const float* hp = h + (size_t)n * (H * C) + hd * C;
  const float* as = att_src + hd * C;
  const float* ad = att_dst + hd * C;
  float s = 0.f, d = 0.f;
#pragma unroll 8
  for (int c = 0; c < C; ++c) {
    float v = hp[c];
    s = fmaf(v, as[c], s);
    d = fmaf(v, ad[c], d);
  }
  asrc[idx] = s;
  adst[idx] = d;
}

// ---------------------------------------------------------------------------
// Per (edge, head): w = exp(leaky_relu(a_src[src]+a_dst[dst], 0.2));
// denom[dst,h] += w.  Max-subtraction omitted (cancels algebraically; logits
// are O(1) so exp cannot overflow; self-loops mean no empty segments).
// Edges e >= Eg are the implicit self-loops (node = e - Eg).
// ---------------------------------------------------------------------------
template <int H>
__global__ void gat_edge_exp(const int* __restrict__ src,
                             const int* __restrict__ dst,
                             int Eg, int N,
                             const float* __restrict__ asrc,
                             const float* __restrict__ adst,
                             float* __restrict__ wexp,
                             float* __restrict__ denom) {
  int idx = blockIdx.x * blockDim.x + threadIdx.x;
  int Et = Eg + N;
  if (idx >= Et * H) return;
  int e  = idx / H, hd = idx & (H - 1);
  int s  = (e < Eg) ? src[e] : (e - Eg);
  int d  = (e < Eg) ? dst[e] : (e - Eg);
  float z = asrc[(size_t)s * H + hd] + adst[(size_t)d * H + hd];
  z = (z > 0.f) ? z : 0.2f * z;
  float w = expf(z);
  wexp[idx] = w;
  atomicAdd(&denom[(size_t)d * H + hd], w);
}

// ---------------------------------------------------------------------------
// One 256-thread block per edge; coalesced read of h[src,:] row and coalesced
// atomic f32 adds into agg[dst,:].
// ---------------------------------------------------------------------------
template <int H, int C>
__global__ void gat_edge_agg(const int* __restrict__ src,
                             const int* __restrict__ dst,
                             int Eg, int N,
                             const float* __restrict__ h,
                             const float* __restrict__ wexp,
                             const float* __restrict__ denom,
                             float* __restrict__ agg) {
  constexpr int HC = H * C;
  int e = blockIdx.x;
  int s = (e < Eg) ? src[e] : (e - Eg);
  int d = (e < Eg) ? dst[e] : (e - Eg);
  const float* hs = h     + (size_t)s * HC;
  float*       o  = agg   + (size_t)d * HC;
  const float* we = wexp  + (size_t)e * H;
  const float* dn = denom + (size_t)d * H;
#pragma unroll
  for (int j = threadIdx.x; j < HC; j += 256) {
    int hd = j / C;                   // C is a power of two -> shift
    float alpha = we[hd] / (dn[hd] + 1e-16f);
    atomicAdd(&o[j], alpha * hs[j]);
  }
}

// out = elu(a + bias) in place (alpha = 1).
template <int COLS>
__global__ void gat_bias_elu(float* __restrict__ a,
                             const float* __restrict__ bias,
                             int total) {
  int i = blockIdx.x * blockDim.x + threadIdx.x;
  if (i >= total) return;
  float v = a[i] + bias[i & (COLS - 1)];
  a[i] = (v > 0.f) ? v : (expf(v) - 1.f);
}

// ---------------------------------------------------------------------------
// One wave per node: mean over HEADS2 heads, + bias, log_softmax across the
// 32 classes (lane == class; wave32 shuffle reductions).
// ---------------------------------------------------------------------------
__global__ __launch_bounds__(32)
void gat_finalize(const float* __restrict__ agg2,
                  const float* __restrict__ bias,
                  float* __restrict__ out, int N) {
  int n = blockIdx.x;
  int c = threadIdx.x;                       // class id, 0..31
  const float* p = agg2 + (size_t)n * (HEADS2 * NCLS);
  float v = 0.f;
#pragma unroll
  for (int hd = 0; hd < HEADS2; ++hd) v += p[hd * NCLS + c];
  v = v * (1.0f / (float)HEADS2) + bias[c];

  float m = v;
#pragma unroll
  for (int off = 16; off > 0; off >>= 1) m = fmaxf(m, __shfl_xor(m, off, 32));
  float ex = expf(v - m);
  float ssum = ex;
#pragma unroll
  for (int off = 16; off > 0; off >>= 1) ssum += __shfl_xor(ssum, off, 32);
  out[(size_t)n * NCLS + c] = (v - m) - logf(ssum);
}

// ---------------------------------------------------------------------------
extern "C" void kernel_launch(void* const* d_in, const int* in_sizes, int n_in,
                              void* d_out, int out_size, void* d_ws, size_t ws_size,
                              hipStream_t stream) {
  (void)n_in; (void)out_size; (void)ws_size;
  const float* x        = (const float*)d_in[0];
  const int*   ei       = (const int*)  d_in[1];
  const float* W1       = (const float*)d_in[2];
  const float* att_src1 = (const float*)d_in[3];
  const float* att_dst1 = (const float*)d_in[4];
  const float* bias1    = (const float*)d_in[5];
  const float* W2       = (const float*)d_in[6];
  const float* att_src2 = (const float*)d_in[7];
  const float* att_dst2 = (const float*)d_in[8];
  const float* bias2    = (const float*)d_in[9];
  float* out = (float*)d_out;

  const int N  = in_sizes[0] / IN_DIM;     // 10000
  const int Eg = in_sizes[1] / 2;          // 50000 given edges
  const int Et = Eg + N;                   // + self loops
  const int* src = ei;
  const int* dst = ei + Eg;

  // ---- workspace layout (floats) ----
  float* ws = (float*)d_ws;
  size_t o = 0;
  float* h1     = ws + o; o += (size_t)N * HC1;   // 40.96M
  float* agg1   = ws + o; o += (size_t)N * HC1;   // 40.96M
  float* asrc1  = ws + o; o += (size_t)N * HEADS1;
  float* adst1  = ws + o; o += (size_t)N * HEADS1;
  float* denom1 = ws + o; o += (size_t)N * HEADS1;
  float* wexp1  = ws + o; o += (size_t)Et * HEADS1;
  // layer-2 buffers alias h1's region: h1 is dead after gat_edge_agg (layer 1)
  size_t o2 = 0;
  float* h2     = h1 + o2; o2 += (size_t)N * HC2;
  float* agg2   = h1 + o2; o2 += (size_t)N * HC2;
  float* asrc2  = h1 + o2; o2 += (size_t)N * HEADS2;
  float* adst2  = h1 + o2; o2 += (size_t)N * HEADS2;
  float* denom2 = h1 + o2; o2 += (size_t)N * HEADS2;
  float* wexp2  = h1 + o2; o2 += (size_t)Et * HEADS2;

  const int TB = 256;

  // ---- layer 1 ----
  hipMemsetAsync(agg1,   0, (size_t)N * HC1 * sizeof(float), stream);
  hipMemsetAsync(denom1, 0, (size_t)N * HEADS1 * sizeof(float), stream);

  dim3 g1(HC1 / 64, (N + 63) / 64);
  gat_wmma_gemm_f32<IN_DIM, HC1><<<g1, 128, GEMM_LDS, stream>>>(x, W1, h1, N);

  gat_att_dots<HEADS1, HID1><<<(N * HEADS1 + TB - 1) / TB, TB, 0, stream>>>(
      h1, att_src1, att_dst1, asrc1, adst1, N);

  gat_edge_exp<HEADS1><<<(Et * HEADS1 + TB - 1) / TB, TB, 0, stream>>>(
      src, dst, Eg, N, asrc1, adst1, wexp1, denom1);

  gat_edge_agg<HEADS1, HID1><<<Et, TB, 0, stream>>>(
      src, dst, Eg, N, h1, wexp1, denom1, agg1);

  gat_bias_elu<HC1><<<((size_t)N * HC1 + TB - 1) / TB, TB, 0, stream>>>(
      agg1, bias1, N * HC1);

  // ---- layer 2 (h1 region now reusable) ----
  hipMemsetAsync(agg2,   0, (size_t)N * HC2 * sizeof(float), stream);
  hipMemsetAsync(denom2, 0, (size_t)N * HEADS2 * sizeof(float), stream);

  dim3 g2(HC2 / 64, (N + 63) / 64);
  gat_wmma_gemm_f32<HC1, HC2><<<g2, 128, GEMM_LDS, stream>>>(agg1, W2, h2, N);

  gat_att_dots<HEADS2, NCLS><<<(N * HEADS2 + TB - 1) / TB, TB, 0, stream>>>(
      h2, att_src2, att_dst2, asrc2, adst2, N);

  gat_edge_exp<HEADS2><<<(Et * HEADS2 + TB - 1) / TB, TB, 0, stream>>>(
      src, dst, Eg, N, asrc2, adst2, wexp2, denom2);

  gat_edge_agg<HEADS2, NCLS><<<Et, TB, 0, stream>>>(
      src, dst, Eg, N, h2, wexp2, denom2, agg2);

  gat_finalize<<<N, 32, 0, stream>>>(agg2, bias2, out, N);
}